// SymptomGNN_29678224015563
// MI455X (gfx1250) — compile-verified
//
#include <hip/hip_runtime.h>
#include <hip/hip_bf16.h>
#include <limits.h>

typedef __attribute__((ext_vector_type(2))) float v2f;
typedef __attribute__((ext_vector_type(8))) float v8f;

#define HID   32
#define HEADS 4
#define FDIM  128   // HEADS*HID
#define NEG_SLOPE 0.2f
#define GEMM_WAVES 8

// ---- order-preserving float<->int encoding for atomicMax on signed int ----
__device__ __forceinline__ int enc_f32(float x) {
  int ix = __float_as_int(x);
  return ix ^ ((ix >> 31) & 0x7FFFFFFF);
}
__device__ __forceinline__ float dec_f32(int e) {
  int ix = e ^ ((e >> 31) & 0x7FFFFFFF);
  return __int_as_float(ix);
}

// =====================================================================
// Input layer: H0[N,32] = relu(X[N,256] @ W[256,32] + b)
// W staged to LDS (repacked so K-pairs are contiguous -> ds_load_b64).
// One wave per 16-row tile; 2 column tiles of 16; V_WMMA_F32_16X16X4_F32.
// =====================================================================
__global__ void __launch_bounds__(256)
in_gemm_relu(const float* __restrict__ X, const float* __restrict__ W,
             const float* __restrict__ bias, float* __restrict__ H, int n) {
  __shared__ float2 wlds[128 * 32];            // (256/2) pairs x 32 cols = 32KB
  int tid = threadIdx.x;
  for (int idx = tid; idx < 256 * 32; idx += 256) {
    int r = idx >> 5, c = idx & 31;            // W[r][c], r = K index
    ((float*)&wlds[(r >> 1) * 32 + c])[r & 1] = W[idx];
  }
  __syncthreads();

  int wave = tid >> 5, lane = tid & 31;
  int row0 = (blockIdx.x * GEMM_WAVES + wave) * 16;
  if (row0 >= n) return;                       // wave-uniform: EXEC all-1s at WMMA
  int m    = lane & 15;
  int kh   = lane >> 4;                        // which K-half (0/1)
  int ncol = lane & 15;
  v8f acc0 = {}, acc1 = {};
  const float* xr = X + (size_t)(row0 + m) * 256;
  for (int k = 0; k < 256; k += 4) {
    v2f a = *(const v2f*)(xr + k + 2 * kh);    // contiguous K pair -> b64
    const float2* wp = wlds + ((k >> 1) + kh) * 32;
    float2 p0 = wp[ncol];
    float2 p1 = wp[16 + ncol];
    v2f b0; b0.x = p0.x; b0.y = p0.y;
    v2f b1; b1.x = p1.x; b1.y = p1.y;
    acc0 = __builtin_amdgcn_wmma_f32_16x16x4_f32(false, a, false, b0, (short)0, acc0, false, false);
    acc1 = __builtin_amdgcn_wmma_f32_16x16x4_f32(false, a, false, b1, (short)0, acc1, false, false);
  }
  float bc0 = bias[ncol], bc1 = bias[16 + ncol];
#pragma unroll
  for (int j = 0; j < 8; ++j) {
    int mm = j + 8 * kh;
    float* hp = H + (size_t)(row0 + mm) * HID;
    hp[ncol]      = fmaxf(acc0[j] + bc0, 0.0f);
    hp[16 + ncol] = fmaxf(acc1[j] + bc1, 0.0f);
  }
}

// =====================================================================
// Layer GEMM: H[N,128] = X[N,din] @ W[din,128]   (din = 32 or 128)
// W staged+repacked in LDS; one wave per 16-row tile, 8 col tiles.
// =====================================================================
__global__ void __launch_bounds__(256)
gat_gemm128(const float* __restrict__ X, const float* __restrict__ W,
            float* __restrict__ H, int n, int din) {
  __shared__ float2 wlds[64 * 128];            // max: (128/2) pairs x 128 = 64KB
  int tid = threadIdx.x;
  int total = din * FDIM;
  for (int idx = tid; idx < total; idx += 256) {
    int r = idx >> 7, c = idx & 127;
    ((float*)&wlds[(r >> 1) * 128 + c])[r & 1] = W[idx];
  }
  __syncthreads();

  int wave = tid >> 5, lane = tid & 31;
  int row0 = (blockIdx.x * GEMM_WAVES + wave) * 16;
  if (row0 >= n) return;
  int m    = lane & 15;
  int kh   = lane >> 4;
  int ncol = lane & 15;
  v8f acc[8] = {};
  const float* xr = X + (size_t)(row0 + m) * din;
  for (int k = 0; k < din; k += 4) {
    v2f a = *(const v2f*)(xr + k + 2 * kh);
    const float2* wp = wlds + ((k >> 1) + kh) * 128;
#pragma unroll
    for (int t = 0; t < 8; ++t) {
      float2 p = wp[t * 16 + ncol];            // ds_load_b64
      v2f b; b.x = p.x; b.y = p.y;
      acc[t] = __builtin_amdgcn_wmma_f32_16x16x4_f32(false, a, false, b, (short)0, acc[t], false, false);
    }
  }
#pragma unroll
  for (int t = 0; t < 8; ++t)
#pragma unroll
    for (int j = 0; j < 8; ++j) {
      int mm = j + 8 * kh;
      H[(size_t)(row0 + mm) * FDIM + t * 16 + ncol] = acc[t][j];
    }
}

// =====================================================================
// Per-node attention coefficients. One wave per node: coalesced float4
// row reads, 8-lane shuffle tree reduction per head.
// =====================================================================
__global__ void attn_coef(const float* __restrict__ H,
                          const float* __restrict__ a_src,
                          const float* __restrict__ a_dst,
                          float* __restrict__ es, float* __restrict__ ed, int n) {
  int gid  = blockIdx.x * blockDim.x + threadIdx.x;
  int node = gid >> 5, lane = gid & 31;
  if (node >= n) return;
  int h = lane >> 3, sub = lane & 7;
  const float4 hv = *(const float4*)(H + (size_t)node * FDIM + lane * 4);
  const float4 av = *(const float4*)(a_src + h * HID + sub * 4);
  const float4 dv = *(const float4*)(a_dst + h * HID + sub * 4);
  float s1 = hv.x * av.x + hv.y * av.y + hv.z * av.z + hv.w * av.w;
  float s2 = hv.x * dv.x + hv.y * dv.y + hv.z * dv.z + hv.w * dv.w;
#pragma unroll
  for (int off = 4; off > 0; off >>= 1) {
    s1 += __shfl_xor(s1, off, 32);
    s2 += __shfl_xor(s2, off, 32);
  }
  if (sub == 0) { es[node * HEADS + h] = s1; ed[node * HEADS + h] = s2; }
}

// =====================================================================
// Zero output accumulator (float4); init segment max (-inf enc), sum (0)
// =====================================================================
__global__ void gat_init(float4* __restrict__ out4, int* __restrict__ mb,
                         float* __restrict__ sb, int n) {
  int i = blockIdx.x * blockDim.x + threadIdx.x;
  if (i < n * (FDIM / 4)) out4[i] = make_float4(0.f, 0.f, 0.f, 0.f);
  if (i < n * HEADS) { mb[i] = INT_MIN; sb[i] = 0.0f; }
}

// =====================================================================
// Edge pass A: e = leaky_relu(e_src[src] + e_dst[dst]); segment-max via
// order-preserving int atomicMax; stash logits in ex.
// =====================================================================
__global__ void edge_max(const int* __restrict__ srci, const int* __restrict__ dsti,
                         int E, int n,
                         const float* __restrict__ es, const float* __restrict__ ed,
                         float* __restrict__ ex, int* __restrict__ mb) {
  int e = blockIdx.x * blockDim.x + threadIdx.x;
  if (e >= E + n) return;
  int s = (e < E) ? srci[e] : (e - E);
  int d = (e < E) ? dsti[e] : (e - E);
#pragma unroll
  for (int h = 0; h < HEADS; ++h) {
    float v = es[s * HEADS + h] + ed[d * HEADS + h];
    v = (v > 0.f) ? v : NEG_SLOPE * v;
    ex[(size_t)e * HEADS + h] = v;
    atomicMax(&mb[d * HEADS + h], enc_f32(v));
  }
}

// =====================================================================
// Edge pass B: ex = exp(e - m[dst]); segment-sum via f32 hardware atomic
// =====================================================================
__global__ void edge_expsum(const int* __restrict__ dsti, int E, int n,
                            float* __restrict__ ex, const int* __restrict__ mb,
                            float* __restrict__ sb) {
  int i = blockIdx.x * blockDim.x + threadIdx.x;   // edge*HEADS + head
  if (i >= (E + n) * HEADS) return;
  int e = i >> 2, h = i & 3;
  int d = (e < E) ? dsti[e] : (e - E);
  float mval = dec_f32(mb[d * HEADS + h]);
  float v = __expf(ex[i] - mval);
  ex[i] = v;
  unsafeAtomicAdd(&sb[d * HEADS + h], v);
}

// =====================================================================
// Edge pass C: out[dst] += h[src] * alpha. One wave per edge; each lane
// owns a float4 of the 128-wide message (8 lanes per head).
// =====================================================================
__global__ void edge_msg(const int* __restrict__ srci, const int* __restrict__ dsti,
                         int E, int n,
                         const float* __restrict__ H, const float* __restrict__ ex,
                         const float* __restrict__ sb, float* __restrict__ out) {
  long long gid = (long long)blockIdx.x * blockDim.x + threadIdx.x;
  int e    = (int)(gid >> 5);
  int lane = (int)(gid & 31);
  if (e >= E + n) return;
  int s = (e < E) ? srci[e] : (e - E);
  int d = (e < E) ? dsti[e] : (e - E);
  int h = lane >> 3;
  float alpha = ex[(size_t)e * HEADS + h] / (sb[d * HEADS + h] + 1e-16f);
  const float4 hv = *(const float4*)(H + (size_t)s * FDIM + lane * 4);
  float* op = out + (size_t)d * FDIM + lane * 4;
  unsafeAtomicAdd(op + 0, hv.x * alpha);
  unsafeAtomicAdd(op + 1, hv.y * alpha);
  unsafeAtomicAdd(op + 2, hv.z * alpha);
  unsafeAtomicAdd(op + 3, hv.w * alpha);
}

// =====================================================================
// out = elu(acc + bias), float4
// =====================================================================
__global__ void bias_elu(const float4* __restrict__ acc, const float* __restrict__ bias,
                         float4* __restrict__ out, int n) {
  int i = blockIdx.x * blockDim.x + threadIdx.x;
  if (i >= n * (FDIM / 4)) return;
  const float4 a = acc[i];
  const float4 bv = ((const float4*)bias)[i & 31];
  float4 v;
  v.x = a.x + bv.x; v.y = a.y + bv.y; v.z = a.z + bv.z; v.w = a.w + bv.w;
  v.x = (v.x > 0.f) ? v.x : (__expf(v.x) - 1.f);
  v.y = (v.y > 0.f) ? v.y : (__expf(v.y) - 1.f);
  v.z = (v.z > 0.f) ? v.z : (__expf(v.z) - 1.f);
  v.w = (v.w > 0.f) ? v.w : (__expf(v.w) - 1.f);
  out[i] = v;
}

extern "C" void kernel_launch(void* const* d_in, const int* in_sizes, int n_in,
                              void* d_out, int out_size, void* d_ws, size_t ws_size,
                              hipStream_t stream) {
  const float* x    = (const float*)d_in[0];
  const int*   ei   = (const int*)d_in[1];
  const float* W_in = (const float*)d_in[2];
  const float* b_in = (const float*)d_in[3];
  const int N    = in_sizes[0] / 256;
  const int E    = in_sizes[1] / 2;
  const int Etot = E + N;

  // workspace layout
  float* bufA = (float*)d_ws;                       // features / output accum [N,128]
  float* bufB = bufA + (size_t)N * FDIM;            // transformed features h  [N,128]
  float* es   = bufB + (size_t)N * FDIM;            // [N,4]
  float* ed   = es + (size_t)N * HEADS;             // [N,4]
  int*   mb   = (int*)(ed + (size_t)N * HEADS);     // [N,4] encoded max
  float* sb   = (float*)(mb + (size_t)N * HEADS);   // [N,4] softmax denom
  float* ex   = sb + (size_t)N * HEADS;             // [Etot,4] logits->exp

  dim3 blk(256);
  int waves = (N + 15) / 16;
  dim3 g_gemm((waves + GEMM_WAVES - 1) / GEMM_WAVES);

  in_gemm_relu<<<g_gemm, blk, 0, stream>>>(x, W_in, b_in, bufA, N);

  for (int l = 0; l < 3; ++l) {
    const float* W  = (const float*)d_in[4 + 4 * l];
    const float* as = (const float*)d_in[5 + 4 * l];
    const float* ad = (const float*)d_in[6 + 4 * l];
    const float* bb = (const float*)d_in[7 + 4 * l];
    int din = (l == 0) ? HID : FDIM;

    gat_gemm128<<<g_gemm, blk, 0, stream>>>(bufA, W, bufB, N, din);
    attn_coef<<<dim3((unsigned)(((long long)N * 32 + 255) / 256)), blk, 0, stream>>>(
        bufB, as, ad, es, ed, N);
    gat_init<<<dim3((N * (FDIM / 4) + 255) / 256), blk, 0, stream>>>(
        (float4*)bufA, mb, sb, N);
    edge_max<<<dim3((Etot + 255) / 256), blk, 0, stream>>>(ei, ei + E, E, N, es, ed, ex, mb);
    edge_expsum<<<dim3((Etot * HEADS + 255) / 256), blk, 0, stream>>>(ei + E, E, N, ex, mb, sb);
    edge_msg<<<dim3((unsigned)(((long long)Etot * 32 + 255) / 256)), blk, 0, stream>>>(
        ei, ei + E, E, N, bufB, ex, sb, bufA);

    float* outp = (l == 2) ? (float*)d_out : bufA;
    bias_elu<<<dim3((N * (FDIM / 4) + 255) / 256), blk, 0, stream>>>(
        (const float4*)bufA, bb, (float4*)outp, N);
  }
}